// PointCLIPStyleBaseline_34892314313415
// MI455X (gfx1250) — compile-verified
//
#include <hip/hip_runtime.h>
#include <math.h>

// ---------------------------------------------------------------------------
// PointCLIP-style pipeline for MI455X (gfx1250, wave32).
//  - render (stats + scatter-max z-buffer): HBM-bound, 2 x 96MB streamed with
//    b128 loads + global_prefetch_b8
//  - conv2/conv3 as implicit GEMM on V_WMMA_F32_16X16X4_F32 (exact f32 math)
//  - conv2 weights staged into LDS via TDM tensor_load_to_lds (+ tensorcnt wait)
//  - small tails (projections / text / logits) scalar
// Workspace use: ~23.7 MB of d_ws (float), laid out below.
// ---------------------------------------------------------------------------

typedef float v2f  __attribute__((ext_vector_type(2)));
typedef float v8f  __attribute__((ext_vector_type(8)));
typedef unsigned int u32x4 __attribute__((ext_vector_type(4)));
typedef int   i32x4 __attribute__((ext_vector_type(4)));
typedef int   i32x8 __attribute__((ext_vector_type(8)));

#define NPTS   262144
#define NBATCH 32
#define NIMG   96            // 32 batches * 3 views

__device__ __forceinline__ float gelu_exact(float x) {
  // torch GELU (exact erf): 0.5*x*(1+erf(x/sqrt(2)))
  return 0.5f * x * (1.0f + erff(x * 0.70710678118654752440f));
}

// ---------------------------------------------------------------------------
// Pass 1: per-batch mean (3) and max-abs scale (1) over the point cloud.
// One block per batch; 4 points (3 x b128) per thread-iteration.
// ---------------------------------------------------------------------------
__device__ float block_reduce(float v, int tid, float* red, int op) {
  __syncthreads();
  red[tid] = v;
  __syncthreads();
  for (int s = 128; s > 0; s >>= 1) {
    if (tid < s) {
      float a = red[tid], b = red[tid + s];
      red[tid] = (op == 0) ? (a + b) : (op == 1 ? fmaxf(a, b) : fminf(a, b));
    }
    __syncthreads();
  }
  return red[0];
}

__global__ __launch_bounds__(256) void stats_kernel(const float* __restrict__ pc,
                                                    float* __restrict__ stats) {
  __shared__ float red[256];
  const int b = blockIdx.x;
  const int tid = threadIdx.x;
  const float4* p4 = (const float4*)(pc + (size_t)b * (NPTS * 3));
  float s[3]  = {0.f, 0.f, 0.f};
  float mx[3] = {-3.4e38f, -3.4e38f, -3.4e38f};
  float mn[3] = { 3.4e38f,  3.4e38f,  3.4e38f};
  for (int i = tid; i < NPTS / 4; i += 256) {
    __builtin_prefetch(p4 + 3 * (i + 2048), 0, 1);   // ~96KB ahead of stream
    const float4 q0 = p4[3 * i + 0];
    const float4 q1 = p4[3 * i + 1];
    const float4 q2 = p4[3 * i + 2];
    const float v[12] = {q0.x, q0.y, q0.z, q0.w, q1.x, q1.y,
                         q1.z, q1.w, q2.x, q2.y, q2.z, q2.w};
#pragma unroll
    for (int j = 0; j < 4; ++j)
#pragma unroll
      for (int c = 0; c < 3; ++c) {
        const float vv = v[j * 3 + c];
        s[c] += vv; mx[c] = fmaxf(mx[c], vv); mn[c] = fminf(mn[c], vv);
      }
  }
  float scale = 0.f, mean[3];
#pragma unroll
  for (int c = 0; c < 3; ++c) {
    float ts  = block_reduce(s[c],  tid, red, 0);
    float tmx = block_reduce(mx[c], tid, red, 1);
    float tmn = block_reduce(mn[c], tid, red, 2);
    mean[c] = ts * (1.0f / (float)NPTS);
    // max |p - mean| over this coord = max(max-mean, mean-min)
    scale = fmaxf(scale, fmaxf(tmx - mean[c], mean[c] - tmn));
  }
  if (tid == 0) {
    stats[b * 4 + 0] = mean[0];
    stats[b * 4 + 1] = mean[1];
    stats[b * 4 + 2] = mean[2];
    stats[b * 4 + 3] = fmaxf(scale, 1e-6f);
  }
}

// ---------------------------------------------------------------------------
// Pass 2: scatter-max z-buffer into 96 (= 32*3) 64x64 images.
// 4 points per thread (3 x b128 loads). dv in [0,1] -> uint atomicMax on
// float bits is order-preserving. Buffer pre-zeroed.
// ---------------------------------------------------------------------------
__global__ __launch_bounds__(256) void scatter_kernel(const float* __restrict__ pc,
                                                      const float* __restrict__ stats,
                                                      unsigned int* __restrict__ buf) {
  const int idx = blockIdx.x * 256 + threadIdx.x;     // over NBATCH*NPTS/4
  if (idx >= NBATCH * (NPTS / 4)) return;
  const int b = idx >> 16;                            // NPTS/4 = 65536
  const float* st = stats + b * 4;
  const float m0 = st[0], m1 = st[1], m2 = st[2];
  const float inv = 1.0f / st[3];
  const float4* p4 = (const float4*)pc + (size_t)idx * 3;
  __builtin_prefetch(p4 + 3 * 2048, 0, 1);
  const float4 q0 = p4[0], q1 = p4[1], q2 = p4[2];
  const float v[12] = {q0.x, q0.y, q0.z, q0.w, q1.x, q1.y,
                       q1.z, q1.w, q2.x, q2.y, q2.z, q2.w};
  unsigned int* ib = buf + (b * 3) * 4096;
#pragma unroll
  for (int j = 0; j < 4; ++j) {
    const float s0 = (v[j * 3 + 0] - m0) * inv;
    const float s1 = (v[j * 3 + 1] - m1) * inv;
    const float s2 = (v[j * 3 + 2] - m2) * inv;
    const float pu[3] = {s0, s1, s0};     // view proj u: (x,y),(y,z),(x,z)
    const float pv[3] = {s1, s2, s2};
    const float dp[3] = {s2, s0, s1};     // depth per view
#pragma unroll
    for (int vI = 0; vI < 3; ++vI) {
      const float L = 63.0f;
      int x = (int)fminf(fmaxf(rintf((pu[vI] + 1.0f) * 0.5f * L), 0.0f), L);
      int y = (int)fminf(fmaxf(rintf((1.0f - (pv[vI] + 1.0f) * 0.5f) * L), 0.0f), L);
      const float dv = (dp[vI] + 1.0f) * 0.5f;          // >= 0
      atomicMax(&ib[vI * 4096 + y * 64 + x], __float_as_uint(dv));
    }
  }
}

// ---------------------------------------------------------------------------
// conv1 (1->16, 3x3, SAME) + exact GELU + 2x2 maxpool, fused. Direct (tiny K).
// out: (96,16,32,32)
// ---------------------------------------------------------------------------
__global__ __launch_bounds__(256) void conv1_kernel(const float* __restrict__ img,
                                                    const float* __restrict__ w,
                                                    const float* __restrict__ bias,
                                                    float* __restrict__ out) {
  const int idx = blockIdx.x * 256 + threadIdx.x;
  if (idx >= NIMG * 16 * 32 * 32) return;
  const int x = idx & 31, y = (idx >> 5) & 31, c = (idx >> 10) & 15, im = idx >> 14;
  const float* I = img + im * 4096;
  float wv[9];
#pragma unroll
  for (int i = 0; i < 9; ++i) wv[i] = w[c * 9 + i];
  const float bv = bias[c];
  float mx = -3.4e38f;
#pragma unroll
  for (int sy = 0; sy < 2; ++sy)
#pragma unroll
    for (int sx = 0; sx < 2; ++sx) {
      const int oy = 2 * y + sy, ox = 2 * x + sx;
      float acc = bv;
#pragma unroll
      for (int dy = 0; dy < 3; ++dy) {
        const int iy = oy + dy - 1;
        if (iy < 0 || iy > 63) continue;
#pragma unroll
        for (int dx = 0; dx < 3; ++dx) {
          const int ix = ox + dx - 1;
          if (ix < 0 || ix > 63) continue;
          acc += I[iy * 64 + ix] * wv[dy * 3 + dx];
        }
      }
      mx = fmaxf(mx, gelu_exact(acc));
    }
  out[idx] = mx;
}

// ---------------------------------------------------------------------------
// conv2 (16->32, K=144) as implicit GEMM on V_WMMA_F32_16X16X4_F32.
// One block per image (8 waves). M=16 contiguous x pixels, N=16 c_out,
// K chunked by 4. Input strip + weights staged in LDS (~58 KB).
// Weights staged via TDM (tensor_load_to_lds, 6-arg toolchain form): 2D
// descriptor, 1 row of 4608 f32 elements, contiguous; completion via
// s_wait_tensorcnt.
// A frag (16x4 f32): lane L holds row M=L&15, K = k0+2*(L>>4), +1.
// B frag (4x16 f32): mirrors A's half-wave interleave (best-guess ordering).
// C/D: lane L, reg r -> (M = r + 8*(L>>4), N = L&15).
// out: gelu(conv+bias), full res (96,32,32,32); pool done separately.
// ---------------------------------------------------------------------------
__global__ __launch_bounds__(256) void conv2_wmma_kernel(const float* __restrict__ in,
                                                         const float* __restrict__ w,
                                                         const float* __restrict__ bias,
                                                         float* __restrict__ out) {
  __shared__ float w_lds[32 * 144];
  __shared__ float b_lds[32];
  __shared__ int   koff[144];          // cin*612 + dy*34 + dx
  __shared__ float in_lds[16 * 18 * 34];
  const int img = blockIdx.x, tid = threadIdx.x;
  const int lane = tid & 31, wave = tid >> 5;
  const int half = lane >> 4, m = lane & 15;

#if __has_builtin(__builtin_amdgcn_tensor_load_to_lds) && \
    __has_builtin(__builtin_amdgcn_s_wait_tensorcnt)
  if (wave == 0) {
    // Tensor DMA descriptor (D#) per CDNA5 ISA.
    const unsigned NE = 32u * 144u;                    // 4608 f32 elements
    const unsigned lds_addr = (unsigned)(size_t)(void*)w_lds;  // addr[31:0] = LDS offset
    const unsigned long long ga = (unsigned long long)(size_t)w;
    u32x4 g0;
    g0[0] = 1u;                                        // count=1, user mode
    g0[1] = lds_addr;                                  // lds_addr[31:0]
    g0[2] = (unsigned)(ga & 0xFFFFFFFFu);              // global_addr[31:0]
    g0[3] = (unsigned)((ga >> 32) & 0x1FFFFFFu) | (2u << 30);  // addr[56:32] | type=2
    i32x8 g1;
    g1[0] = (int)(2u << 16);                           // wg_mask=0, data_size=4B
    g1[1] = (int)((NE & 0xFFFFu) << 16);               // tensor_dim0[15:0]
    g1[2] = (int)((NE >> 16) | (1u << 16));            // tensor_dim0[31:16], tensor_dim1=1
    g1[3] = (int)((NE & 0xFFFFu) << 16);               // tile_dim0 = NE
    g1[4] = 1;                                         // tile_dim1 = 1, tile_dim2 = 0
    g1[5] = (int)NE;                                   // tensor_dim0_stride[31:0]
    g1[6] = (int)((NE & 0xFFFFu) << 16);               // dim0_str[47:32]=0, dim1_str[15:0]
    g1[7] = 0;                                         // dim1_stride[47:16]
    i32x4 g2 = {0, 0, 0, 0}, g3 = {0, 0, 0, 0};
    i32x8 g4 = {0, 0, 0, 0, 0, 0, 0, 0};               // 6-arg form: extra group
    __builtin_amdgcn_tensor_load_to_lds(g0, g1, g2, g3, g4, 0);
    __builtin_amdgcn_s_wait_tensorcnt(0);              // data in LDS before barrier
  }
#else
  for (int i = tid; i < 32 * 144; i += 256) w_lds[i] = w[i];
#endif
  if (tid < 32) b_lds[tid] = bias[tid];
  if (tid < 144) {
    int cin = tid / 9, r9 = tid % 9;
    koff[tid] = cin * (18 * 34) + (r9 / 3) * 34 + (r9 % 3);
  }

  const float* inimg = in + img * (16 * 1024);
  float* outimg = out + img * (32 * 1024);
  // wave-constant tile coords: t = wave + 8*ti
  const int nbase = (wave & 1) * 16;
  const int x0 = ((wave >> 1) & 1) * 16;

  for (int strip = 0; strip < 2; ++strip) {
    const int y0 = strip * 16;
    __syncthreads();
    for (int i = tid; i < 16 * 18 * 34; i += 256) {
      int cin = i / 612, rem = i % 612, yy = rem / 34, xx = rem % 34;
      int gy = y0 - 1 + yy, gx = xx - 1;
      in_lds[i] = (gy >= 0 && gy < 32 && gx >= 0 && gx < 32)
                      ? inimg[cin * 1024 + gy * 32 + gx] : 0.0f;
    }
    __syncthreads();

    const float bval = b_lds[nbase + m];
    const float* wrow = &w_lds[(nbase + m) * 144];
    for (int ti = 0; ti < 8; ++ti) {
      const int yl = (wave >> 2) + 2 * ti;       // local output row 0..15
      const int baseA = yl * 34 + x0 + m;
      v8f c;
#pragma unroll
      for (int r = 0; r < 8; ++r) c[r] = bval;   // bias broadcast: N = lane&15
#pragma unroll 4
      for (int k0 = 0; k0 < 144; k0 += 4) {
        const int kA = k0 + 2 * half;
        v2f a, bb;
        a[0]  = in_lds[baseA + koff[kA]];
        a[1]  = in_lds[baseA + koff[kA + 1]];
        bb[0] = wrow[kA];
        bb[1] = wrow[kA + 1];
        c = __builtin_amdgcn_wmma_f32_16x16x4_f32(false, a, false, bb,
                                                  (short)0, c, false, false);
      }
      const int gy = y0 + yl;
      float* op = outimg + (nbase + m) * 1024 + gy * 32 + x0 + 8 * half;
#pragma unroll
      for (int r = 0; r < 8; ++r) op[r] = gelu_exact(c[r]);
    }
  }
}

// 2x2 maxpool after conv2: (96,32,32,32) -> (96,32,16,16)
__global__ void pool2x2_kernel(const float* __restrict__ in, float* __restrict__ out) {
  const int idx = blockIdx.x * 256 + threadIdx.x;
  if (idx >= NIMG * 32 * 16 * 16) return;
  const int x = idx & 15, y = (idx >> 4) & 15, c = (idx >> 8) & 31, im = idx >> 13;
  const float* p = in + ((im * 32 + c) * 32 + 2 * y) * 32 + 2 * x;
  out[idx] = fmaxf(fmaxf(p[0], p[1]), fmaxf(p[32], p[33]));
}

// ---------------------------------------------------------------------------
// conv3 (32->64, K=288) WMMA + exact GELU + global average pool -> (96,64).
// One block per image. Input staged in LDS (32x18x18 padded, ~42 KB);
// weights streamed from L2. Per-wave register accumulation + exactly two
// commutative LDS float adds per feature cell (bit-deterministic).
// ---------------------------------------------------------------------------
__global__ __launch_bounds__(256) void conv3_wmma_kernel(const float* __restrict__ in,
                                                         const float* __restrict__ w,
                                                         const float* __restrict__ bias,
                                                         float* __restrict__ feat) {
  __shared__ float in_lds[32 * 18 * 18];
  __shared__ int   koff[288];          // cin*324 + dy*18 + dx
  __shared__ float feat_lds[64];
  const int img = blockIdx.x, tid = threadIdx.x;
  const int lane = tid & 31, wave = tid >> 5;
  const int half = lane >> 4, m = lane & 15;

  if (tid < 64) feat_lds[tid] = 0.0f;
  for (int i = tid; i < 288; i += 256) {
    int cin = i / 9, r9 = i % 9;
    koff[i] = cin * 324 + (r9 / 3) * 18 + (r9 % 3);
  }
  const float* inimg = in + img * (32 * 256);
  for (int i = tid; i < 32 * 18 * 18; i += 256) {
    int cin = i / 324, rem = i % 324, yy = rem / 18, xx = rem % 18;
    int gy = yy - 1, gx = xx - 1;
    in_lds[i] = (gy >= 0 && gy < 16 && gx >= 0 && gx < 16)
                    ? inimg[cin * 256 + gy * 16 + gx] : 0.0f;
  }
  __syncthreads();

  const int nbase = (wave & 3) * 16;   // t&3 is wave-constant (t = wave + 8*ti)
  const float bval = bias[nbase + m];
  const float* wrow = w + (nbase + m) * 288;
  float ssum = 0.0f;
  for (int ti = 0; ti < 8; ++ti) {
    const int yl = (wave >> 2) + 2 * ti;       // output row 0..15, M = x
    const int baseA = yl * 18 + m;
    v8f c;
#pragma unroll
    for (int r = 0; r < 8; ++r) c[r] = bval;
#pragma unroll 4
    for (int k0 = 0; k0 < 288; k0 += 4) {
      const int kA = k0 + 2 * half;
      v2f a, bb;
      a[0]  = in_lds[baseA + koff[kA]];
      a[1]  = in_lds[baseA + koff[kA + 1]];
      bb[0] = wrow[kA];
      bb[1] = wrow[kA + 1];
      c = __builtin_amdgcn_wmma_f32_16x16x4_f32(false, a, false, bb,
                                                (short)0, c, false, false);
    }
    float s = 0.0f;
#pragma unroll
    for (int r = 0; r < 8; ++r) s += gelu_exact(c[r]);   // gelu then pixel-sum
    s += __shfl_xor(s, 16, 32);        // combine the two x-halves (same N)
    ssum += s;                          // rows accumulate in registers
  }
  if (half == 0) atomicAdd(&feat_lds[nbase + m], ssum);  // 2 adds/cell total
  __syncthreads();
  if (tid < 64) feat[img * 64 + tid] = feat_lds[tid] * (1.0f / 256.0f);
}

// ---------------------------------------------------------------------------
// View projection (64->64) + mean over 3 views + L2-norm. Block per batch.
// ---------------------------------------------------------------------------
__global__ __launch_bounds__(64) void viewproj_kernel(const float* __restrict__ feat,
                                                      const float* __restrict__ pw,
                                                      const float* __restrict__ pb,
                                                      float* __restrict__ vf) {
  __shared__ float f_lds[3 * 64];
  __shared__ float red[64];
  const int b = blockIdx.x, o = threadIdx.x;
  for (int v = 0; v < 3; ++v) f_lds[v * 64 + o] = feat[(b * 3 + v) * 64 + o];
  __syncthreads();
  float e = 0.0f;
  for (int v = 0; v < 3; ++v) {
    float acc = 0.0f;
    for (int c = 0; c < 64; ++c) acc += f_lds[v * 64 + c] * pw[o * 64 + c];
    e += acc;
  }
  e = e * (1.0f / 3.0f) + pb[o];
  red[o] = e * e;
  __syncthreads();
  for (int s = 32; s > 0; s >>= 1) { if (o < s) red[o] += red[o + s]; __syncthreads(); }
  const float nrm = fmaxf(sqrtf(red[0]), 1e-12f);
  vf[b * 64 + o] = e / nrm;
}

// ---------------------------------------------------------------------------
// Text encoder: embedding masked-mean + projection + L2-norm. Block per class.
// ---------------------------------------------------------------------------
__global__ __launch_bounds__(64) void text_kernel(const int* __restrict__ tok,
                                                  const float* __restrict__ table,
                                                  const float* __restrict__ tw,
                                                  const float* __restrict__ tb,
                                                  float* __restrict__ tf) {
  __shared__ float pooled[64];
  __shared__ float red[64];
  const int cls = blockIdx.x, d = threadIdx.x;
  float s = 0.0f; int cnt = 0;
  for (int l = 0; l < 48; ++l) {
    const int id = tok[cls * 48 + l];
    if (id != 0) { s += table[id * 64 + d]; ++cnt; }
  }
  pooled[d] = s / fmaxf((float)cnt, 1.0f);
  __syncthreads();
  float e = tb[d];
  for (int c = 0; c < 64; ++c) e += pooled[c] * tw[d * 64 + c];
  red[d] = e * e;
  __syncthreads();
  for (int st = 32; st > 0; st >>= 1) { if (d < st) red[d] += red[d + st]; __syncthreads(); }
  const float nrm = fmaxf(sqrtf(red[0]), 1e-12f);
  tf[cls * 64 + d] = e / nrm;
}

// Final logits: exp(logit_scale) * vf @ tf.T -> (32,40)
__global__ void logits_kernel(const float* __restrict__ vf, const float* __restrict__ tf,
                              const float* __restrict__ ls, float* __restrict__ out) {
  const int idx = blockIdx.x * 256 + threadIdx.x;
  if (idx >= NBATCH * 40) return;
  const int c = idx % 40, b = idx / 40;
  float acc = 0.0f;
  for (int o = 0; o < 64; ++o) acc += vf[b * 64 + o] * tf[c * 64 + o];
  out[idx] = expf(ls[0]) * acc;
}

// ---------------------------------------------------------------------------
extern "C" void kernel_launch(void* const* d_in, const int* in_sizes, int n_in,
                              void* d_out, int out_size, void* d_ws, size_t ws_size,
                              hipStream_t stream) {
  const float* pc  = (const float*)d_in[0];
  const float* w1  = (const float*)d_in[1];
  const float* b1  = (const float*)d_in[2];
  const float* w2  = (const float*)d_in[3];
  const float* b2  = (const float*)d_in[4];
  const float* w3  = (const float*)d_in[5];
  const float* b3  = (const float*)d_in[6];
  const float* pw  = (const float*)d_in[7];
  const float* pb  = (const float*)d_in[8];
  const float* emb = (const float*)d_in[9];
  const float* tw  = (const float*)d_in[10];
  const float* tb  = (const float*)d_in[11];
  const float* ls  = (const float*)d_in[12];
  const int*   tok = (const int*)d_in[13];
  float* out = (float*)d_out;

  // Workspace layout (floats); total ~5.91M floats (~23.7 MB)
  float* ws    = (float*)d_ws;
  float* stats = ws;                 // 32*4
  float* imgs  = ws + 128;           // 96*4096
  float* c1o   = ws + 393344;        // 96*16*32*32
  float* c2f   = ws + 1966208;       // 96*32*32*32
  float* c2p   = ws + 5111936;       // 96*32*16*16
  float* feat  = ws + 5898368;       // 96*64
  float* vf    = ws + 5904512;       // 32*64
  float* tf    = ws + 5906560;       // 40*64

  (void)hipMemsetAsync(imgs, 0, (size_t)NIMG * 4096 * sizeof(float), stream);
  stats_kernel<<<NBATCH, 256, 0, stream>>>(pc, stats);
  scatter_kernel<<<(NBATCH * (NPTS / 4) + 255) / 256, 256, 0, stream>>>(
      pc, stats, (unsigned int*)imgs);
  conv1_kernel<<<(NIMG * 16 * 32 * 32 + 255) / 256, 256, 0, stream>>>(imgs, w1, b1, c1o);
  conv2_wmma_kernel<<<NIMG, 256, 0, stream>>>(c1o, w2, b2, c2f);
  pool2x2_kernel<<<(NIMG * 32 * 16 * 16 + 255) / 256, 256, 0, stream>>>(c2f, c2p);
  conv3_wmma_kernel<<<NIMG, 256, 0, stream>>>(c2p, w3, b3, feat);
  viewproj_kernel<<<NBATCH, 64, 0, stream>>>(feat, pw, pb, vf);
  text_kernel<<<40, 64, 0, stream>>>(tok, emb, tw, tb, tf);
  logits_kernel<<<(NBATCH * 40 + 255) / 256, 256, 0, stream>>>(vf, tf, ls, out);
}